// BayesianGaussianMixture_20444044329723
// MI455X (gfx1250) — compile-verified
//
#include <hip/hip_runtime.h>

#define NPTS 32768
#define DIM  64
#define KCMP 64

typedef float v2f __attribute__((ext_vector_type(2)));
typedef float v8f __attribute__((ext_vector_type(8)));

// ---------------------------------------------------------------- digamma ----
__device__ __forceinline__ float digammaf_dev(float x) {
    // valid for x > 0 (all call sites guarantee this)
    float r = 0.0f;
    while (x < 6.0f) { r -= 1.0f / x; x += 1.0f; }
    float inv  = 1.0f / x;
    float inv2 = inv * inv;
    float s = logf(x) - 0.5f * inv
            - inv2 * (1.0f/12.0f - inv2 * (1.0f/120.0f - inv2 * (1.0f/252.0f)));
    return r + s;
}

// ---------------------------------------------------------- setup kernel 1 ---
// muP[k,d] = sum_e means[k,e] * P[k,e,d]
__global__ void bgm_muP_kernel(const float* __restrict__ means,
                               const float* __restrict__ P,
                               float* __restrict__ muP) {
    int k = blockIdx.x;
    int d = threadIdx.x;
    const float* mk = means + k * DIM;
    const float* Pk = P + k * DIM * DIM + d;
    float s = 0.0f;
    #pragma unroll 8
    for (int e = 0; e < DIM; ++e)
        s = fmaf(mk[e], Pk[e * DIM], s);
    muP[k * DIM + d] = s;
}

// ---------------------------------------------------------- setup kernel 2 ---
// constk[k] = -0.5*D*log(2pi) + log_det_k - 0.5*D*log(dof_k)
//           + 0.5*(log_lambda_k - D/mp_k) + log_w_k
__global__ void bgm_const_kernel(const float* __restrict__ P,
                                 const float* __restrict__ wc,   // [2,K]
                                 const float* __restrict__ dof,  // [K]
                                 const float* __restrict__ mp,   // [K]
                                 float* __restrict__ constk) {
    __shared__ float scan[KCMP];
    int k = threadIdx.x;

    const float* Pk = P + k * DIM * DIM;
    float logdet = 0.0f;
    for (int d = 0; d < DIM; ++d)
        logdet += logf(Pk[d * DIM + d]);

    float dk = dof[k];
    float loglam = (float)DIM * 0.6931471805599453f; // D*log(2)
    for (int i = 0; i < DIM; ++i)
        loglam += digammaf_dev(0.5f * (dk - (float)i));

    float a = wc[k];            // weight_concentration[0][k]
    float b = wc[KCMP + k];     // weight_concentration[1][k]
    float dsum = digammaf_dev(a + b);
    float lw_local = digammaf_dev(a) - dsum;
    float cterm    = digammaf_dev(b) - dsum;

    scan[k] = cterm;
    __syncthreads();
    float excl = 0.0f;
    for (int j = 0; j < k; ++j) excl += scan[j];

    const float log2pi = 1.8378770664093453f;
    float ck = -0.5f * (float)DIM * log2pi
             + logdet
             - 0.5f * (float)DIM * logf(dk)
             + 0.5f * (loglam - (float)DIM / mp[k])
             + lw_local + excl;
    constk[k] = ck;
}

// --------------------------------------------------------- lane reduction ----
// Merging tree: 8 per-lane partials (rows r / r+8) summed over each 16-lane
// half using 8 shuffles total. Result: lane L holds the full sum for row
// (L&7) + 8*(L>>4), duplicated at lane L^8.
__device__ __forceinline__ float merge_red(float a, float b, int mask, int lane) {
    const bool hi = (lane & mask) != 0;
    float keep = hi ? b : a;
    float send = hi ? a : b;
    return keep + __shfl_xor(send, mask);
}

__device__ __forceinline__ float reduce8x16(const float s[8], int lane) {
    float v01 = merge_red(s[0], s[1], 1, lane);
    float v23 = merge_red(s[2], s[3], 1, lane);
    float v45 = merge_red(s[4], s[5], 1, lane);
    float v67 = merge_red(s[6], s[7], 1, lane);
    float v03 = merge_red(v01, v23, 2, lane);
    float v47 = merge_red(v45, v67, 2, lane);
    float v   = merge_red(v03, v47, 4, lane);
    v += __shfl_xor(v, 8);
    return v;
}

// ------------------------------------------------------------- main kernel ---
// One wave handles TWO 16-row tiles of X (32 rows); loops over all K
// components. Each B operand is one bank-conflict-free ds_load_b64 thanks to
// a pair-interleaved, half-swizzled LDS layout; each B feeds two WMMAs.
__global__ void __launch_bounds__(256)
bgm_argmax_kernel(const float* __restrict__ X,
                  const float* __restrict__ P,
                  const float* __restrict__ muP,
                  const float* __restrict__ constk,
                  int* __restrict__ out) {
    // Pair-interleaved P_k: element (d,e) at dword
    //   (d>>1)*128 + ((2e) ^ ((d>>1 & 1)<<5)) + (d&1)
    __shared__ float sP2[DIM * DIM];  // 16 KB
    __shared__ float sMu[DIM];
    __shared__ float sC[1];

    const int tid  = threadIdx.x;
    const int wave = tid >> 5;
    const int lane = tid & 31;
    const int m    = lane & 15;   // A row within half / B column index
    const int half = lane >> 4;   // 0: K-pair {0,1}; 1: K-pair {2,3}

    const int rowBase = (blockIdx.x * 8 + wave) * 32;

    // ---- Load two 16x64 X tiles into A-operand registers.
    // A (16x4 f32): lane m holds row m; VGPR0=K(2*half), VGPR1=K(2*half+1).
    v2f a0[16], a1[16];
    const float* x0 = X + (size_t)(rowBase      + m) * DIM;
    const float* x1 = X + (size_t)(rowBase + 16 + m) * DIM;
    #pragma unroll
    for (int c = 0; c < 16; ++c) {
        a0[c] = *(const v2f*)(x0 + 4 * c + 2 * half);
        a1[c] = *(const v2f*)(x1 + 4 * c + 2 * half);
    }

    float best0 = -3.402823466e+38f, best1 = -3.402823466e+38f;
    int   bk0 = 0, bk1 = 0;

    for (int k = 0; k < KCMP; ++k) {
        __syncthreads();   // previous iteration's LDS reads done

        // ---- Stage P_k into LDS, pair-interleaved (2 float4 loads + 4 b64
        // stores per 2x4 sub-block; 2 sub-blocks per thread).
        const float* src = P + (size_t)k * DIM * DIM;
        #pragma unroll
        for (int i = 0; i < 2; ++i) {
            const int blk   = tid + 256 * i;       // 0..511
            const int dpair = blk >> 4;            // 0..31
            const int e4    = (blk & 15) * 4;      // 0..60
            const float4 r0 = *(const float4*)(src + (2 * dpair)     * DIM + e4);
            const float4 r1 = *(const float4*)(src + (2 * dpair + 1) * DIM + e4);
            const int sw = (dpair & 1) << 5;
            float* dst = sP2 + dpair * 128;
            v2f p;
            p.x = r0.x; p.y = r1.x; *(v2f*)(dst + ((2 * (e4 + 0)) ^ sw)) = p;
            p.x = r0.y; p.y = r1.y; *(v2f*)(dst + ((2 * (e4 + 1)) ^ sw)) = p;
            p.x = r0.z; p.y = r1.z; *(v2f*)(dst + ((2 * (e4 + 2)) ^ sw)) = p;
            p.x = r0.w; p.y = r1.w; *(v2f*)(dst + ((2 * (e4 + 3)) ^ sw)) = p;
        }
        if (tid < DIM) sMu[tid] = muP[k * DIM + tid];
        if (tid == 0)  sC[0] = constk[k];
        // pull next component's P tile toward the caches while we compute
        if (k + 1 < KCMP)
            __builtin_prefetch(P + (size_t)(k + 1) * DIM * DIM + tid * 16, 0, 0);
        __syncthreads();

        float sq0[8], sq1[8];
        #pragma unroll
        for (int r = 0; r < 8; ++r) { sq0[r] = 0.0f; sq1[r] = 0.0f; }

        #pragma unroll
        for (int g = 0; g < 4; ++g) {
            const int col = 16 * g + m;
            // lane-constant base; per-c offset is a 256-dword (1024-byte) immediate.
            // d0 = 4c + 2*half -> dpair = 2c + half, parity = half.
            const float* bp = sP2 + 128 * half + ((2 * col) ^ (half << 5));
            v8f acc0 = {};
            v8f acc1 = {};
            #pragma unroll
            for (int c = 0; c < 16; ++c) {
                const v2f b = *(const v2f*)(bp + 256 * c);  // one ds_load_b64
                acc0 = __builtin_amdgcn_wmma_f32_16x16x4_f32(
                        false, a0[c], false, b, (short)0, acc0, false, false);
                acc1 = __builtin_amdgcn_wmma_f32_16x16x4_f32(
                        false, a1[c], false, b, (short)0, acc1, false, false);
            }
            const float mu = sMu[col];
            #pragma unroll
            for (int r = 0; r < 8; ++r) {
                float t0 = acc0[r] - mu;
                sq0[r] = fmaf(t0, t0, sq0[r]);
                float t1 = acc1[r] - mu;
                sq1[r] = fmaf(t1, t1, sq1[r]);
            }
        }

        // ---- Column reduction: lane L ends with row (L&7)+8*half.
        const float s0 = reduce8x16(sq0, lane);
        const float s1 = reduce8x16(sq1, lane);

        const float ck = sC[0];
        const float w0 = fmaf(-0.5f, s0, ck);
        const float w1 = fmaf(-0.5f, s1, ck);
        const bool p0 = w0 > best0;   // strict: first max wins (jnp.argmax)
        best0 = p0 ? w0 : best0;
        bk0   = p0 ? k  : bk0;
        const bool p1 = w1 > best1;
        best1 = p1 ? w1 : best1;
        bk1   = p1 ? k  : bk1;
    }

    if ((lane & 8) == 0) {
        const int row = (lane & 7) + 8 * half;
        out[rowBase      + row] = bk0;
        out[rowBase + 16 + row] = bk1;
    }
}

// ------------------------------------------------------------------ launch ---
extern "C" void kernel_launch(void* const* d_in, const int* in_sizes, int n_in,
                              void* d_out, int out_size, void* d_ws, size_t ws_size,
                              hipStream_t stream) {
    const float* X     = (const float*)d_in[0];  // [N,D]
    const float* means = (const float*)d_in[1];  // [K,D]
    const float* P     = (const float*)d_in[2];  // [K,D,D]
    const float* wc    = (const float*)d_in[3];  // [2,K]
    const float* dof   = (const float*)d_in[4];  // [K]
    const float* mp    = (const float*)d_in[5];  // [K]
    int* out = (int*)d_out;                      // argmax indices (int32)

    float* muP    = (float*)d_ws;                // K*D floats
    float* constk = muP + KCMP * DIM;            // K floats

    bgm_muP_kernel  <<<KCMP, DIM, 0, stream>>>(means, P, muP);
    bgm_const_kernel<<<1,    KCMP, 0, stream>>>(P, wc, dof, mp, constk);
    bgm_argmax_kernel<<<NPTS / (32 * 8), 256, 0, stream>>>(X, P, muP, constk, out);
}